// MM_47717086659307
// MI455X (gfx1250) — compile-verified
//
#include <hip/hip_runtime.h>
#include <hip/hip_bf16.h>

// Sizes fixed by the reference
#define DIM   50
#define LS    5
#define NNEG  100
#define NC    110     // pos(5) + neg(100) + seq(5) candidate rows of the Gram matrix
#define NCAND 105     // pos + neg  (sim candidates / output width)
#define NT    7       // N tiles: 7*16 = 112 >= 110
#define NK    13      // K steps:  13*4 = 52 >= 50
#define WAVES 8       // waves (batch elements) per block

typedef float v2f __attribute__((ext_vector_type(2)));
typedef float v8f __attribute__((ext_vector_type(8)));

__device__ __forceinline__ float wred_max(float v) {
#pragma unroll
  for (int o = 16; o > 0; o >>= 1) v = fmaxf(v, __shfl_xor(v, o, 32));
  return v;
}
__device__ __forceinline__ float wred_sum(float v) {
#pragma unroll
  for (int o = 16; o > 0; o >>= 1) v += __shfl_xor(v, o, 32);
  return v;
}

__global__ __launch_bounds__(32 * WAVES)
void sasrec_fused_kernel(const int* __restrict__ seqs,
                         const int* __restrict__ pos,
                         const int* __restrict__ neg,
                         const float* __restrict__ emb_item,
                         float* __restrict__ out, int B)
{
  __shared__ float ldsS[WAVES][LS][NT * 16]; // Gram rows S[s][n], n in [0,112)
  __shared__ float ldsW[WAVES][LS][104];     // weights rows (101 used)

  const int lane = threadIdx.x & 31;
  const int wv   = threadIdx.x >> 5;
  long b = (long)blockIdx.x * WAVES + wv;
  if (b >= B) b = B - 1;                     // duplicate last element: deterministic, no barrier skew

  const int   row   = lane & 15;             // M-row (A) / N-col-in-tile (B)
  const int   hi    = lane >> 4;             // lane half selects K pair
  const float scale = 0.14142135623730950488f; // 1/sqrt(50)

  float (*S)[NT * 16] = ldsS[wv];
  float (*W)[104]     = ldsW[wv];

#if __has_builtin(__builtin_amdgcn_wmma_f32_16x16x4_f32)
  // ---------------- WMMA path: S = A(16x52) * B(52x112), f32 16x16x4 ----------------
  // A fragment (kk): vgpr j = seq[row][4kk + 2*hi + j]   (rows >=5 / cols >=50 are zero)
  v2f afrag[NK];
  {
    const int  sidx = (row < LS) ? seqs[b * LS + row] : 0;
    const float* ar = emb_item + (size_t)sidx * DIM;
#pragma unroll
    for (int kk = 0; kk < NK; kk++) {
      int col = 4 * kk + 2 * hi;
      v2f a = {0.0f, 0.0f};
      if (row < LS && col < DIM) a = *(const v2f*)(ar + col);   // 8B aligned (col even)
      afrag[kk] = a;
    }
  }

#pragma unroll
  for (int t = 0; t < NT; t++) {
    const int n = 16 * t + row;
    int cidx = 0;
    if      (n < LS)    cidx = pos[b * LS + n];
    else if (n < NCAND) cidx = neg[b * NNEG + (n - LS)];
    else if (n < NC)    cidx = seqs[b * LS + (n - NCAND)];
    const bool nvalid = (n < NC);
    const float* br = emb_item + (size_t)cidx * DIM;

    // B fragment (kk): vgpr j = cand[n][4kk + 2*hi + j]  (same K striping as A)
    v2f bfrag[NK];
#pragma unroll
    for (int kk = 0; kk < NK; kk++) {
      int col = 4 * kk + 2 * hi;
      v2f bf = {0.0f, 0.0f};
      if (nvalid && col < DIM) bf = *(const v2f*)(br + col);
      bfrag[kk] = bf;
    }

    v8f acc = {0.f, 0.f, 0.f, 0.f, 0.f, 0.f, 0.f, 0.f};
#pragma unroll
    for (int kk = 0; kk < NK; kk++) {
      acc = __builtin_amdgcn_wmma_f32_16x16x4_f32(
          /*neg_a=*/false, afrag[kk], /*neg_b=*/false, bfrag[kk],
          /*c_mod=*/(short)0, acc, /*reuse_a=*/false, /*reuse_b=*/false);
    }
    // D layout: vgpr j, lanes 0-15 -> M=j ; only rows 0..4 are meaningful
    if (hi == 0) {
#pragma unroll
      for (int j = 0; j < LS; j++) S[j][16 * t + row] = acc[j];
    }
  }
#else
  // ---------------- VALU fallback (keeps kernel compiling if builtin missing) ------
  for (int i = lane; i < LS * DIM; i += 32)
    W[i / DIM][i % DIM] = emb_item[(size_t)seqs[b * LS + i / DIM] * DIM + (i % DIM)];
  __syncthreads();
  for (int n = lane; n < NT * 16; n += 32) {
    int cidx = 0;
    if      (n < LS)    cidx = pos[b * LS + n];
    else if (n < NCAND) cidx = neg[b * NNEG + (n - LS)];
    else if (n < NC)    cidx = seqs[b * LS + (n - NCAND)];
    const float* cr = emb_item + (size_t)cidx * DIM;
    float d[LS] = {0.f, 0.f, 0.f, 0.f, 0.f};
    if (n < NC)
      for (int k = 0; k < DIM; k++) {
        float cv = cr[k];
#pragma unroll
        for (int s = 0; s < LS; s++) d[s] += W[s][k] * cv;
      }
#pragma unroll
    for (int s = 0; s < LS; s++) S[s][n] = d[s];
  }
#endif
  __syncthreads();

  // ---- attention softmax (over 5) + weights, all from the Gram matrix -------------
  // logits:  c<5 : l[sp] = (sp==c) ? 0 : scale*S[c][105+sp]   (masked diagonal)
  //          c>=5: l[sp] = scale*S[sp][c]
  // w[s][0]     = scale * sum_sp attn[s][sp]   * S[s][105+sp]      (w_seq)
  // w[s][c-4]   = scale * sum_sp attn[c][sp]   * S[s][105+sp]      (w_neg, c in [5,105))
  for (int c = lane; c < NCAND; c += 32) {
    float lg[LS];
    float m = -3.4e38f;
#pragma unroll
    for (int sp = 0; sp < LS; sp++) {
      float l = (c < LS) ? ((sp == c) ? 0.0f : scale * S[c][NCAND + sp])
                         : scale * S[sp][c];
      lg[sp] = l;
      m = fmaxf(m, l);
    }
    float den = 0.f;
#pragma unroll
    for (int sp = 0; sp < LS; sp++) { lg[sp] = expf(lg[sp] - m); den += lg[sp]; }
    float inv = 1.0f / den;
#pragma unroll
    for (int s = 0; s < LS; s++) {
      float rd = 0.f;
#pragma unroll
      for (int sp = 0; sp < LS; sp++) rd += lg[sp] * S[s][NCAND + sp];
      rd *= scale * inv;
      if (c < LS) { if (s == c) W[c][0] = rd; }
      else        W[s][c - 4] = rd;
    }
  }
  __syncthreads();

  // ---- per-(b,s): ctx = log_softmax(w)[0]; out = log_softmax(sim) + ctx ----------
#pragma unroll
  for (int s = 0; s < LS; s++) {
    float m = -3.4e38f;
    for (int k = lane; k < 101; k += 32) m = fmaxf(m, W[s][k]);
    m = wred_max(m);
    float se = 0.f;
    for (int k = lane; k < 101; k += 32) se += expf(W[s][k] - m);
    se = wred_sum(se);
    float ctx = W[s][0] - (m + logf(se));

    float m2 = -3.4e38f;
    for (int c = lane; c < NCAND; c += 32) m2 = fmaxf(m2, scale * S[s][c]);
    m2 = wred_max(m2);
    float se2 = 0.f;
    for (int c = lane; c < NCAND; c += 32) se2 += expf(scale * S[s][c] - m2);
    se2 = wred_sum(se2);
    float off = ctx - (m2 + logf(se2));

    float* op = out + ((size_t)b * LS + s) * NCAND;
    for (int c = lane; c < NCAND; c += 32) op[c] = scale * S[s][c] + off;
  }
}

extern "C" void kernel_launch(void* const* d_in, const int* in_sizes, int n_in,
                              void* d_out, int out_size, void* d_ws, size_t ws_size,
                              hipStream_t stream) {
  (void)n_in; (void)out_size; (void)d_ws; (void)ws_size;
  // inputs: 0 users (unused), 1 sequences, 2 pos_targets, 3 neg_targets,
  //         4 emb_item, 5 emb_user (unused)
  const int*   seqs     = (const int*)d_in[1];
  const int*   pos      = (const int*)d_in[2];
  const int*   neg      = (const int*)d_in[3];
  const float* emb_item = (const float*)d_in[4];
  float*       out      = (float*)d_out;

  const int B = in_sizes[0];                  // 16384
  const int blocks = (B + WAVES - 1) / WAVES; // one wave32 per batch element
  sasrec_fused_kernel<<<blocks, 32 * WAVES, 0, stream>>>(seqs, pos, neg, emb_item, out, B);
}